// KalmanFilter_41618233098284
// MI455X (gfx1250) — compile-verified
//
#include <hip/hip_runtime.h>

// Batched Kalman filter: B=16384 batches, D=64.
// One 256-thread workgroup (8 wave32 waves) per batch.
// All 64x64x64 matmuls via V_WMMA_F32_16X16X4_F32 (FP32 matrix path on gfx1250).
// Q/R fragments preloaded into registers at stage 0 so HBM latency is hidden
// behind the first two WMMA matmul stages.

#define LDSD 68   // LDS row stride in floats (64 + 4 pad: bank-conflict-free, 16B-aligned rows)

typedef __attribute__((ext_vector_type(2))) float v2f;
typedef __attribute__((ext_vector_type(4))) float f4;
typedef __attribute__((ext_vector_type(8))) float v8f;

// Compute one 16x16 output tile of C(64x64) = Aop(64x64) x Bop(64x64) (+acc).
// Aop/Bop live in LDS with row stride LDSD. If TRANSB, Bop[k][n] = B[n][k].
template <bool TRANSB>
__device__ __forceinline__ v8f wmma_tile64(const float* __restrict__ Aop,
                                           const float* __restrict__ Bop,
                                           int ti, int tj, int lmod, int lhalf,
                                           v8f acc)
{
    const int arow = ti * 16 + lmod;       // A fragment: lane 0-15 -> M rows
    const int bcol = tj * 16 + lmod;       // B fragment: lane 0-15 -> N cols
    const int koff = 2 * lhalf;            // lane half selects K 0/1 vs 2/3
#pragma unroll
    for (int kk = 0; kk < 16; ++kk) {
        const int k0 = kk * 4;
        v2f a, b;
        a.x = Aop[arow * LDSD + k0 + koff];
        a.y = Aop[arow * LDSD + k0 + koff + 1];
        if (TRANSB) {
            b.x = Bop[bcol * LDSD + k0 + koff];
            b.y = Bop[bcol * LDSD + k0 + koff + 1];
        } else {
            b.x = Bop[(k0 + koff) * LDSD + bcol];
            b.y = Bop[(k0 + koff + 1) * LDSD + bcol];
        }
        // D = A(16x4) x B(4x16) + C  -> v_wmma_f32_16x16x4_f32
        acc = __builtin_amdgcn_wmma_f32_16x16x4_f32(
            /*neg_a=*/false, a, /*neg_b=*/false, b,
            /*c_mod=*/(short)0, acc, /*reuse_a=*/false, /*reuse_b=*/false);
    }
    return acc;
}

__global__ __launch_bounds__(256) void kalman64_kernel(
    const float* __restrict__ gx,   // prev_latent_states [B,64]
    const float* __restrict__ gP,   // prev_state_cov     [B,64,64]
    const float* __restrict__ gA,   // transition_A       [B,64,64]
    const float* __restrict__ gQ,   // process_noise      [B,64,64]
    const float* __restrict__ gz,   // observation        [B,64]
    const float* __restrict__ gR,   // observation_noise  [B,64,64]
    float* __restrict__ out_x,      // [B,64]
    float* __restrict__ out_P)      // [B,64,64]
{
    __shared__ float ldsA[64 * LDSD];   // A -> Pc
    __shared__ float ldsB[64 * LDSD];   // P -> T -> S -> Sinv -> K
    __shared__ float sx[64], sz[64], spred[64], sinnov[64];
    __shared__ float prow[64], pcol[64];

    const int tid   = threadIdx.x;
    const int lane  = tid & 31;
    const int lmod  = lane & 15;
    const int lhalf = lane >> 4;
    const int wave  = tid >> 5;          // 0..7
    // tile assignment: 2 tiles per wave over the 4x4 tile grid (same tile-row i)
    const int t0 = wave * 2, t1 = wave * 2 + 1;
    const int i0 = t0 >> 2, j0 = t0 & 3;
    const int i1 = t1 >> 2, j1 = t1 & 3;

    // fragment element coordinates per (tile, VGPR v): m = i*16 + v + 8*lhalf, n = j*16 + lmod
    const int fm0 = i0 * 16 + 8 * lhalf, fn0 = j0 * 16 + lmod;
    const int fm1 = i1 * 16 + 8 * lhalf, fn1 = j1 * 16 + lmod;

    const size_t b    = blockIdx.x;
    const size_t mb   = b * 4096;
    const float* A_g  = gA + mb;
    const float* P_g  = gP + mb;
    const float* Q_g  = gQ + mb;
    const float* R_g  = gR + mb;

    // ---- Stage 0: load A, P into LDS (float4), x/z vectors, and issue the
    //      Q/R fragment loads early so stages 1-2 hide their latency ----
#pragma unroll
    for (int t = 0; t < 4; ++t) {
        const int idx4 = tid + t * 256;          // 0..1023 float4s
        const int row  = idx4 >> 4;
        const int c    = (idx4 & 15) * 4;
        f4 va = *(const f4*)(A_g + row * 64 + c);
        f4 vp = *(const f4*)(P_g + row * 64 + c);
        *(f4*)(&ldsA[row * LDSD + c]) = va;
        *(f4*)(&ldsB[row * LDSD + c]) = vp;
    }
    v8f accQ0, accQ1, fragR0, fragR1;
#pragma unroll
    for (int v = 0; v < 8; ++v) {
        accQ0[v]  = Q_g[(fm0 + v) * 64 + fn0];
        accQ1[v]  = Q_g[(fm1 + v) * 64 + fn1];
        fragR0[v] = R_g[(fm0 + v) * 64 + fn0];
        fragR1[v] = R_g[(fm1 + v) * 64 + fn1];
    }
    if (tid < 64) {
        sx[tid] = gx[b * 64 + tid];
        sz[tid] = gz[b * 64 + tid];
    }
    __syncthreads();

    // ---- Stage 1: predicted_state = A x ; innovation = z - pred ----
    if (tid < 64) {
        float acc = 0.f;
#pragma unroll 8
        for (int c = 0; c < 64; ++c) acc += ldsA[tid * LDSD + c] * sx[c];
        spred[tid]  = acc;
        sinnov[tid] = sz[tid] - acc;
    }
    __syncthreads();

    // ---- Stage 2: T = A @ P  (regs) ----
    v8f accT0 = {}; v8f accT1 = {};
    accT0 = wmma_tile64<false>(ldsA, ldsB, i0, j0, lmod, lhalf, accT0);
    accT1 = wmma_tile64<false>(ldsA, ldsB, i1, j1, lmod, lhalf, accT1);
    __syncthreads();
    // T overwrites P's buffer
#pragma unroll
    for (int v = 0; v < 8; ++v) {
        ldsB[(fm0 + v) * LDSD + fn0] = accT0[v];
        ldsB[(fm1 + v) * LDSD + fn1] = accT1[v];
    }
    __syncthreads();

    // ---- Stage 3: Pc = T @ A^T + Q  (acc preloaded with Q fragments) ----
    accQ0 = wmma_tile64<true>(ldsB, ldsA, i0, j0, lmod, lhalf, accQ0);
    accQ1 = wmma_tile64<true>(ldsB, ldsA, i1, j1, lmod, lhalf, accQ1);
    __syncthreads();
    // Pc overwrites A's buffer; S = Pc + R overwrites T's buffer
#pragma unroll
    for (int v = 0; v < 8; ++v) {
        ldsA[(fm0 + v) * LDSD + fn0] = accQ0[v];
        ldsA[(fm1 + v) * LDSD + fn1] = accQ1[v];
        ldsB[(fm0 + v) * LDSD + fn0] = accQ0[v] + fragR0[v];
        ldsB[(fm1 + v) * LDSD + fn1] = accQ1[v] + fragR1[v];
    }
    __syncthreads();

    // ---- Stage 4: in-place Gauss-Jordan inversion of S (SPD, no pivoting) ----
    for (int k = 0; k < 64; ++k) {
        const float d  = ldsB[k * LDSD + k];
        const float id = 1.0f / d;
        if (tid < 64) {
            prow[tid] = ldsB[k * LDSD + tid] * id;   // new pivot row
            pcol[tid] = ldsB[tid * LDSD + k];        // old pivot column
        }
        __syncthreads();
#pragma unroll
        for (int t = 0; t < 16; ++t) {
            const int idx = tid + t * 256;           // 0..4095
            const int i = idx >> 6, j = idx & 63;
            float val;
            if (i == k)      val = (j == k) ? id : prow[j];
            else if (j == k) val = -pcol[i] * id;
            else             val = ldsB[i * LDSD + j] - pcol[i] * prow[j];
            ldsB[i * LDSD + j] = val;
        }
        __syncthreads();
    }

    // ---- Stage 5: K = Pc @ Sinv ----
    v8f accK0 = {}; v8f accK1 = {};
    accK0 = wmma_tile64<false>(ldsA, ldsB, i0, j0, lmod, lhalf, accK0);
    accK1 = wmma_tile64<false>(ldsA, ldsB, i1, j1, lmod, lhalf, accK1);
    __syncthreads();
    // K overwrites Sinv's buffer
#pragma unroll
    for (int v = 0; v < 8; ++v) {
        ldsB[(fm0 + v) * LDSD + fn0] = accK0[v];
        ldsB[(fm1 + v) * LDSD + fn1] = accK1[v];
    }
    __syncthreads();

    // ---- Stage 6: x_new = pred + K @ innovation ----
    if (tid < 64) {
        float acc = spred[tid];
#pragma unroll 8
        for (int c = 0; c < 64; ++c) acc += ldsB[tid * LDSD + c] * sinnov[c];
        out_x[b * 64 + tid] = acc;
    }

    // ---- Stage 7: P_new = Pc - K @ Pc  (== Pc - K S K^T for exact Sinv) ----
    v8f accM0 = {}; v8f accM1 = {};
    accM0 = wmma_tile64<false>(ldsB, ldsA, i0, j0, lmod, lhalf, accM0);
    accM1 = wmma_tile64<false>(ldsB, ldsA, i1, j1, lmod, lhalf, accM1);
#pragma unroll
    for (int v = 0; v < 8; ++v) {
        out_P[mb + (fm0 + v) * 64 + fn0] = ldsA[(fm0 + v) * LDSD + fn0] - accM0[v];
        out_P[mb + (fm1 + v) * 64 + fn1] = ldsA[(fm1 + v) * LDSD + fn1] - accM1[v];
    }
}

extern "C" void kernel_launch(void* const* d_in, const int* in_sizes, int n_in,
                              void* d_out, int out_size, void* d_ws, size_t ws_size,
                              hipStream_t stream) {
    const float* x = (const float*)d_in[0];  // prev_latent_states [B,64]
    const float* P = (const float*)d_in[1];  // prev_state_cov     [B,64,64]
    const float* A = (const float*)d_in[2];  // transition_A       [B,64,64]
    const float* Q = (const float*)d_in[3];  // process_noise      [B,64,64]
    const float* z = (const float*)d_in[4];  // observation        [B,64]
    const float* R = (const float*)d_in[5];  // observation_noise  [B,64,64]

    const int B = in_sizes[0] / 64;
    float* out_x = (float*)d_out;                    // [B,64]
    float* out_P = (float*)d_out + (size_t)B * 64;   // [B,64,64]

    kalman64_kernel<<<B, 256, 0, stream>>>(x, P, A, Q, z, R, out_x, out_P);
}